// LinearAttention_36498632082002
// MI455X (gfx1250) — compile-verified
//
#include <hip/hip_runtime.h>
#include <hip/hip_bf16.h>

typedef float v2f __attribute__((ext_vector_type(2)));
typedef float v8f __attribute__((ext_vector_type(8)));

#define BH   64
#define LL   1024
#define DK   1024
#define DV   64
#define EPSF 1e-6f

__device__ __forceinline__ float elup(float x) {
    // elu(x) + 1  (alpha = 1)
    return x > 0.0f ? x + 1.0f : __expf(x);
}

__global__ __launch_bounds__(256) void la_zero_kernel(float* __restrict__ p, int n) {
    int i = blockIdx.x * 256 + threadIdx.x;
    if (i < n) p[i] = 0.0f;
}

// Pass 1: row sums of elu(Q)+1 and elu(K)+1, plus column sums of elu(K)+1.
// grid.x = BH*8 (8 blocks of 128 rows each), grid.y = 0 (Q) / 1 (K).
// Block = 256 threads = 8 wave32; wave w owns rows rb*128 + w*16 .. +15.
__global__ __launch_bounds__(256) void la_reduce_kernel(
    const float* __restrict__ Qg, const float* __restrict__ Kg,
    float* __restrict__ Qrs, float* __restrict__ Krs, float* __restrict__ Kcs)
{
    __shared__ float stage[8 * 16 * 34];  // per-wave 16x32 partials, stride 34 (pad)
    __shared__ float scol[DK];            // block-level column accumulator

    const int bh   = blockIdx.x >> 3;
    const int rb   = blockIdx.x & 7;
    const int mat  = blockIdx.y;          // 0 = Q, 1 = K
    const int tid  = threadIdx.x;
    const int wave = tid >> 5;
    const int lane = tid & 31;

    for (int i = tid; i < DK; i += 256) scol[i] = 0.0f;
    __syncthreads();

    const float* __restrict__ src = (mat ? Kg : Qg) + (size_t)bh * LL * DK;
    const int rowbase = rb * 128 + wave * 16;

    float colacc[32];
    #pragma unroll
    for (int i = 0; i < 32; ++i) colacc[i] = 0.0f;

    // Sweep: lane j covers columns {128*i + 4*j .. +3}, i = 0..7, per row.
    float acc[16];
    #pragma unroll
    for (int r = 0; r < 16; ++r) {
        const float4* __restrict__ rp =
            (const float4*)(src + (size_t)(rowbase + r) * DK) + lane;
        float s = 0.0f;
        #pragma unroll
        for (int i = 0; i < 8; ++i) {
            float4 v = rp[i * 32];
            float e0 = elup(v.x), e1 = elup(v.y), e2 = elup(v.z), e3 = elup(v.w);
            s += (e0 + e1) + (e2 + e3);
            if (mat) {
                colacc[i * 4 + 0] += e0; colacc[i * 4 + 1] += e1;
                colacc[i * 4 + 2] += e2; colacc[i * 4 + 3] += e3;
            }
        }
        acc[r] = s;
    }

    // Stage P[m][lane] = acc[m] for the cross-lane WMMA reduction.
    float* __restrict__ st = &stage[wave * 16 * 34];
    #pragma unroll
    for (int m = 0; m < 16; ++m) st[m * 34 + lane] = acc[m];

    if (mat) {
        #pragma unroll
        for (int i = 0; i < 8; ++i)
            #pragma unroll
            for (int cc = 0; cc < 4; ++cc)
                atomicAdd(&scol[i * 128 + lane * 4 + cc], colacc[i * 4 + cc]);
    }
    __syncthreads();

    // Row-sum reduce in exact fp32: D = P(16x32) x ones(32x16)
    // via 8 chained V_WMMA_F32_16X16X4_F32.
    // ISA A-layout (16x4 f32): lane m<16 holds {A[m,k0], A[m,k0+1]},
    // lane m+16 holds {A[m,k0+2], A[m,k0+3]}  -> v2f LDS fetch below.
    const int m  = lane & 15;
    const int hi = lane >> 4;
    v2f bones = {1.0f, 1.0f};   // B = ones: layout-invariant
    v8f cacc = {};
    #pragma unroll
    for (int j0 = 0; j0 < 32; j0 += 4) {
        const v2f a = *(const v2f*)&st[m * 34 + j0 + hi * 2];
        cacc = __builtin_amdgcn_wmma_f32_16x16x4_f32(
            false, a, false, bones, (short)0, cacc, false, false);
    }

    // D layout: VGPR i, lanes 0-15 -> M=i, lanes 16-31 -> M=8+i (all N equal).
    float* __restrict__ rs = (mat ? Krs : Qrs) + bh * LL + rowbase;
    if (lane == 0) {
        #pragma unroll
        for (int i = 0; i < 8; ++i) rs[i] = cacc[i];
    } else if (lane == 16) {
        #pragma unroll
        for (int i = 0; i < 8; ++i) rs[8 + i] = cacc[i];
    }

    // Flush block-level column sums of K to global.
    if (mat) {
        float* __restrict__ kc = Kcs + bh * DK;
        #pragma unroll
        for (int cc = 0; cc < 4; ++cc) {
            int col = tid * 4 + cc;
            atomicAdd(&kc[col], scol[col]);
        }
    }
}

// Pass 2: out[bh,l,v] = V[bh,l,v] * Qrs*Krs / (Qrs*Kcs + eps), float2 per thread.
__global__ __launch_bounds__(256) void la_scale_kernel(
    const float* __restrict__ Vg, const float* __restrict__ Qrs,
    const float* __restrict__ Krs, const float* __restrict__ Kcs,
    float* __restrict__ out)
{
    int t   = blockIdx.x * 256 + threadIdx.x;  // over BH*LL*(DV/2)
    int row = t >> 5;                          // DV/2 = 32 float2 per row
    int c2  = t & 31;
    float q  = Qrs[row];
    float k  = Krs[row];
    float kc = Kcs[row];                       // [bh][k==l], same 1024 stride
    float scale = q * k / (q * kc + EPSF);
    const float2* __restrict__ v2p = (const float2*)Vg;
    float2 v = v2p[(size_t)row * 32 + c2];
    float2 o; o.x = v.x * scale; o.y = v.y * scale;
    ((float2*)out)[(size_t)row * 32 + c2] = o;
}

extern "C" void kernel_launch(void* const* d_in, const int* in_sizes, int n_in,
                              void* d_out, int out_size, void* d_ws, size_t ws_size,
                              hipStream_t stream) {
    const float* Q = (const float*)d_in[0];
    const float* K = (const float*)d_in[1];
    const float* V = (const float*)d_in[2];
    float* out = (float*)d_out;

    float* Qrs = (float*)d_ws;         // BH*LL floats
    float* Krs = Qrs + BH * LL;        // BH*LL floats
    float* Kcs = Krs + BH * LL;        // BH*DK floats
    (void)in_sizes; (void)n_in; (void)out_size; (void)ws_size;

    la_zero_kernel<<<(BH * DK) / 256, 256, 0, stream>>>(Kcs, BH * DK);

    dim3 g1(BH * 8, 2);
    la_reduce_kernel<<<g1, 256, 0, stream>>>(Q, K, Qrs, Krs, Kcs);

    int n2 = BH * LL * (DV / 2);
    la_scale_kernel<<<n2 / 256, 256, 0, stream>>>(V, Qrs, Krs, Kcs, out);
}